// MixedPoolingLayer_84172769068198
// MI455X (gfx1250) — compile-verified
//
#include <hip/hip_runtime.h>

// Mixed 2x2 pooling (training mode): out = (1-k)*max(window) + k*mean(window)
// x: [32, 256, 256, 128] f32 (NHWC, C innermost), k: [32, 128, 128, 128] i32 in {0,1}
// out: [32, 128, 128, 128] f32.
//
// Pure HBM-bandwidth-bound streaming kernel (~1.61 GB @ 23.3 TB/s => ~69 us floor).
// Strategy: one thread = one float4 of output channels. Lanes of a wave32 cover
// the 128-channel dim (32 lanes x float4 = 512 B = one pixel) so every VMEM op
// is a fully coalesced 128-bit access. Everything is single-use, so all traffic
// carries non-temporal cache hints (gfx1250 TH_NT).

typedef float v4f __attribute__((ext_vector_type(4)));
typedef int   v4i __attribute__((ext_vector_type(4)));

namespace {
constexpr unsigned kC4   = 32;           // 128 channels / 4 floats per vec
constexpr unsigned kWo   = 128;          // output width
constexpr unsigned kHo   = 128;          // output height
constexpr unsigned kWin  = 256;          // input width
constexpr unsigned kRow4 = kWin * kC4;   // float4 elements per input row (8192)
}

__global__ __launch_bounds__(256) void mixed_pool2x2_nt(
    const float* __restrict__ xf, const int* __restrict__ ki,
    float* __restrict__ outf, unsigned n4)
{
  unsigned v = blockIdx.x * blockDim.x + threadIdx.x;
  if (v >= n4) return;

  // Decompose flat float4-output index: v = ((b*Ho + ho)*Wo + wo)*C4 + c4
  unsigned c4  = v & (kC4 - 1);
  unsigned pix = v >> 5;                 // /32
  unsigned wo  = pix & (kWo - 1);
  unsigned t   = pix >> 7;               // /128
  unsigned ho  = t & (kHo - 1);
  unsigned b   = t >> 7;                 // /128

  // Base float4 index of x[b, 2*ho, 2*wo, 4*c4]; window neighbors are at
  // +kC4 (next pixel, +512B) and +kRow4 (next row, +128KiB) — both fit the
  // signed 24-bit instruction offset, so one address feeds all four loads.
  unsigned base = ((b * 256u + 2u * ho) * kWin + 2u * wo) * kC4 + c4;

  const v4f* __restrict__ x = (const v4f*)xf;
  v4f x00 = __builtin_nontemporal_load(x + base);
  v4f x01 = __builtin_nontemporal_load(x + base + kC4);
  v4f x10 = __builtin_nontemporal_load(x + base + kRow4);
  v4f x11 = __builtin_nontemporal_load(x + base + kRow4 + kC4);
  v4i kv  = __builtin_nontemporal_load((const v4i*)ki + v);

  v4f r;
#pragma unroll
  for (int i = 0; i < 4; ++i) {
    float mx = fmaxf(fmaxf(x00[i], x01[i]), fmaxf(x10[i], x11[i]));
    float mn = ((x00[i] + x01[i]) + (x10[i] + x11[i])) * 0.25f;
    // k is exactly 0 or 1, so the affine blend is an exact select.
    r[i] = (kv[i] != 0) ? mn : mx;
  }
  __builtin_nontemporal_store(r, (v4f*)outf + v);
}

extern "C" void kernel_launch(void* const* d_in, const int* in_sizes, int n_in,
                              void* d_out, int out_size, void* d_ws, size_t ws_size,
                              hipStream_t stream) {
  const float* x = (const float*)d_in[0];   // [32,256,256,128] f32
  const int*   k = (const int*)d_in[1];     // [32,128,128,128] i32
  float*     out = (float*)d_out;           // [32,128,128,128] f32

  unsigned n4 = (unsigned)(out_size / 4);   // 16,777,216 float4 outputs
  constexpr unsigned block = 256;           // 8 wave32 per block
  unsigned grid = (n4 + block - 1) / block; // 65,536 blocks
  mixed_pool2x2_nt<<<grid, block, 0, stream>>>(x, k, out, n4);
}